// FrameEnergyLoss_63788854280708
// MI455X (gfx1250) — compile-verified
//
#include <hip/hip_runtime.h>
#include <hip/hip_bf16.h>

typedef __attribute__((ext_vector_type(2))) float v2f;
typedef __attribute__((ext_vector_type(8))) float v8f;

#define K1_BLOCKS 1024
#define K2_BLOCKS 512
#define RED_THREADS 256

// ---------------------------------------------------------------------------
// Kernel 1: per-element internal energy, block-reduced partial sums.
// ---------------------------------------------------------------------------
__global__ __launch_bounds__(RED_THREADS)
void elem_energy_kernel(const float* __restrict__ pred_raw,
                        const int*   __restrict__ connectivity,
                        const float* __restrict__ elem_lengths,
                        const float* __restrict__ prop_E,
                        const float* __restrict__ prop_A,
                        const float* __restrict__ prop_I22,
                        const float* __restrict__ elem_dirs,
                        const float* __restrict__ u_c_p,
                        const float* __restrict__ theta_c_p,
                        float* __restrict__ partials,
                        int E)
{
    const float uc = u_c_p[0];
    const float tc = theta_c_p[0];

    float acc = 0.0f;
    const int tid    = threadIdx.x;
    const int stride = gridDim.x * blockDim.x;
    for (int e = blockIdx.x * blockDim.x + tid; e < E; e += stride) {
        // connectivity is (E,2) int32 -> 8-byte aligned int2 per element
        const int2 nn = ((const int2*)connectivity)[e];
        const int nA = nn.x, nB = nn.y;

        // gather node displacements (L2-resident, 12 MB)
        const float pA0 = pred_raw[3 * nA + 0];
        const float pA1 = pred_raw[3 * nA + 1];
        const float pA2 = pred_raw[3 * nA + 2];
        const float pB0 = pred_raw[3 * nB + 0];
        const float pB1 = pred_raw[3 * nB + 1];
        const float pB2 = pred_raw[3 * nB + 2];

        const float uAx = pA0 * uc, uAy = pA1 * uc, thA = pA2 * tc;
        const float uBx = pB0 * uc, uBy = pB1 * uc, thB = pB2 * tc;

        const float c = elem_dirs[3 * e + 0];
        const float s = elem_dirs[3 * e + 2];

        const float L    = elem_lengths[e];
        const float Ee   = prop_E[e];
        const float EA   = Ee * prop_A[e];
        const float EI   = Ee * prop_I22[e];
        const float invL = 1.0f / L;

        const float u_A =  c * uAx + s * uAy;
        const float w_A = -s * uAx + c * uAy;
        const float u_B =  c * uBx + s * uBy;
        const float w_B = -s * uBx + c * uBy;

        const float du = u_B - u_A;
        const float dw = w_B - w_A;

        const float U_ax = 0.5f * (EA * invL) * du * du;
        // w_A^2 + w_B^2 - 2 w_A w_B == dw^2
        const float bend = 4.0f * thA * thA + 4.0f * thB * thB + 4.0f * thA * thB
                         + 12.0f * invL * invL * dw * dw
                         - 12.0f * invL * dw * (thA + thB);
        const float U_bd = EI * 0.5f * invL * bend;

        acc += U_ax + U_bd;
    }

    __shared__ float red[RED_THREADS];
    red[tid] = acc;
    __syncthreads();
    for (int off = RED_THREADS / 2; off > 0; off >>= 1) {
        if (tid < off) red[tid] += red[tid + off];
        __syncthreads();
    }
    if (tid == 0) partials[blockIdx.x] = red[0];
}

// ---------------------------------------------------------------------------
// Kernel 2: -W_external = -sum(F_ext * u_phys), block-reduced partial sums.
// ---------------------------------------------------------------------------
__global__ __launch_bounds__(RED_THREADS)
void ext_work_kernel(const float* __restrict__ pred_raw,
                     const float* __restrict__ F_ext,
                     const float* __restrict__ u_c_p,
                     const float* __restrict__ theta_c_p,
                     float* __restrict__ partials,
                     int N)
{
    const float uc = u_c_p[0];
    const float tc = theta_c_p[0];

    float acc = 0.0f;
    const int tid    = threadIdx.x;
    const int stride = gridDim.x * blockDim.x;
    for (int n = blockIdx.x * blockDim.x + tid; n < N; n += stride) {
        const float p0 = pred_raw[3 * n + 0];
        const float p1 = pred_raw[3 * n + 1];
        const float p2 = pred_raw[3 * n + 2];
        const float f0 = F_ext[3 * n + 0];
        const float f1 = F_ext[3 * n + 1];
        const float f2 = F_ext[3 * n + 2];
        acc -= uc * (p0 * f0 + p1 * f1) + tc * (p2 * f2);
    }

    __shared__ float red[RED_THREADS];
    red[tid] = acc;
    __syncthreads();
    for (int off = RED_THREADS / 2; off > 0; off >>= 1) {
        if (tid < off) red[tid] += red[tid + off];
        __syncthreads();
    }
    if (tid == 0) partials[blockIdx.x] = red[0];
}

// ---------------------------------------------------------------------------
// Kernel 3: single-wave final reduction of all partials using
// V_WMMA_F32_16X16X4_F32 for the cross-lane sum (exact: B = ones).
//   A layout (16x4 f32): lanes 0-15 -> {K0,K1}, lanes 16-31 -> {K2,K3},
//   with a.y = 0 we get D[m][n] = acc[m] + acc[m+16] for all n.
//   Lane l<16 holds D[r][l] in VGPR r (r=0..7); lane l>=16 holds D[8+r][l-16].
//   total = (sum of 8 regs in lane g) + (sum in lane g^16).
// ---------------------------------------------------------------------------
__global__ __launch_bounds__(32)
void final_reduce_kernel(const float* __restrict__ partials, int P,
                         const float* __restrict__ u_c_p,
                         const float* __restrict__ F_c_p,
                         float* __restrict__ out)
{
    const int lane = threadIdx.x;

    float acc = 0.0f;
    for (int i = lane; i < P; i += 32) acc += partials[i];

#if __has_builtin(__builtin_amdgcn_wmma_f32_16x16x4_f32)
    v2f a; a.x = acc;  a.y = 0.0f;
    v2f b; b.x = 1.0f; b.y = 1.0f;
    v8f cm = {};
    v8f d = __builtin_amdgcn_wmma_f32_16x16x4_f32(
        /*neg_a=*/false, a, /*neg_b=*/false, b,
        /*c_mod=*/(short)0, cm, /*reuse_a=*/false, /*reuse_b=*/false);
    float s = d[0] + d[1] + d[2] + d[3] + d[4] + d[5] + d[6] + d[7];
    float total = s + __shfl_xor(s, 16, 32);
#else
    float total = acc;
    for (int off = 16; off > 0; off >>= 1) total += __shfl_xor(total, off, 32);
#endif

    if (lane == 0) {
        const float E_c = fmaxf(F_c_p[0] * u_c_p[0], 1e-30f);
        out[0] = total / E_c;
    }
}

// ---------------------------------------------------------------------------
extern "C" void kernel_launch(void* const* d_in, const int* in_sizes, int n_in,
                              void* d_out, int out_size, void* d_ws, size_t ws_size,
                              hipStream_t stream)
{
    const float* pred_raw     = (const float*)d_in[0];
    const int*   connectivity = (const int*)  d_in[1];
    const float* elem_lengths = (const float*)d_in[2];
    const float* prop_E       = (const float*)d_in[3];
    const float* prop_A       = (const float*)d_in[4];
    const float* prop_I22     = (const float*)d_in[5];
    const float* elem_dirs    = (const float*)d_in[6];
    const float* F_ext        = (const float*)d_in[7];
    const float* u_c          = (const float*)d_in[8];
    const float* theta_c      = (const float*)d_in[9];
    const float* F_c          = (const float*)d_in[10];

    const int E = in_sizes[2];       // elem_lengths has E entries
    const int N = in_sizes[0] / 3;   // pred_raw is (N,3)

    float* part1 = (float*)d_ws;                 // K1_BLOCKS partials (U_internal)
    float* part2 = part1 + K1_BLOCKS;            // K2_BLOCKS partials (-W_external)

    elem_energy_kernel<<<K1_BLOCKS, RED_THREADS, 0, stream>>>(
        pred_raw, connectivity, elem_lengths, prop_E, prop_A, prop_I22,
        elem_dirs, u_c, theta_c, part1, E);

    ext_work_kernel<<<K2_BLOCKS, RED_THREADS, 0, stream>>>(
        pred_raw, F_ext, u_c, theta_c, part2, N);

    final_reduce_kernel<<<1, 32, 0, stream>>>(
        part1, K1_BLOCKS + K2_BLOCKS, u_c, F_c, (float*)d_out);
}